// HungarianMatcher_64793876627942
// MI455X (gfx1250) — compile-verified
//
#include <hip/hip_runtime.h>
#include <hip/hip_bf16.h>

typedef __attribute__((ext_vector_type(2))) float v2f;
typedef __attribute__((ext_vector_type(8))) float v8f;

#define NUM_KP 17

// One wave computes a 16(query) x 16(target) tile of the cost matrix.
// sq(n,t) per keypoint j is computed with V_WMMA_F32_16X16X4_F32 via:
//   A row n = [1, px, py, px^2+py^2],  B col t = [tx^2+ty^2, -2tx, -2ty, 1]
// f32 16x16x4 fragment layout (ISA 7.12.2):
//   A: lanes 0-15 hold M=0..15 {K0,K1}; lanes 16-31 hold M=0..15 {K2,K3}
//   B: lanes 0-15 hold N=0..15 {K0,K1}; lanes 16-31 hold N=0..15 {K2,K3}
//   D: VGPR i, lane l -> M = i + 8*(l>>4), N = l&15
__global__ __launch_bounds__(640) void matcher_cost_kernel(
    const float* __restrict__ pred_logits,   // (N, 2)
    const float* __restrict__ pred_kp,       // (N, 51)
    const float* __restrict__ tgt_kp,        // (T, 51)
    const float* __restrict__ tgt_area,      // (T,)
    const int*   __restrict__ tgt_ids,       // (T,)
    float* __restrict__ out,                 // (N, T)
    int T)
{
    __shared__ float s_pred[16 * 34];   // 16 query rows x (x,y interleaved 34)
    __shared__ float s_cc[16 * 2];      // focal (pos-neg) per query row, per class

    const int n0  = blockIdx.x * 16;
    const int tid = threadIdx.x;

    // Stage the 16-query keypoint tile into LDS.
    if (tid < 16 * 34) {
        const int r = tid / 34;
        const int k = tid - r * 34;
        s_pred[tid] = pred_kp[(size_t)(n0 + r) * (3 * NUM_KP) + k];
    } else if (tid < 16 * 34 + 32) {
        // Focal class cost for the 16 queries x 2 classes.
        const int u = tid - 16 * 34;
        const int r = u >> 1, c = u & 1;
        const float x = pred_logits[(size_t)(n0 + r) * 2 + c];
        const float p = 1.0f / (1.0f + __expf(-x));
        const float pos = 0.25f * (1.0f - p) * (1.0f - p) * (-__logf(p + 1e-8f));
        const float neg = 0.75f * p * p * (-__logf(1.0f - p + 1e-8f));
        s_cc[u] = pos - neg;
    }
    __syncthreads();

    const int lane = tid & 31;
    const int wave = tid >> 5;
    const int half = lane >> 4;   // 0: lanes 0-15, 1: lanes 16-31
    const int q    = lane & 15;   // A-row source index AND D column index
    const int t    = wave * 16 + q;

    // Per-lane target data (kept in registers; L2-resident loads).
    const float* trow = tgt_kp + (size_t)t * (3 * NUM_KP);
    float tx[NUM_KP], ty[NUM_KP], Vv[NUM_KP];
    float vsum = 0.0f;
    #pragma unroll
    for (int j = 0; j < NUM_KP; ++j) {
        tx[j] = trow[2 * j];
        ty[j] = trow[2 * j + 1];
        Vv[j] = trow[2 * NUM_KP + j];
        vsum += Vv[j];
    }
    const float inv_area = 1.0f / tgt_area[t];
    const int   cls      = tgt_ids[t];
    const float inv_vs   = 1.0f / (vsum + 1e-6f);

    float okacc[8], kptacc[8];
    #pragma unroll
    for (int i = 0; i < 8; ++i) { okacc[i] = 0.0f; kptacc[i] = 0.0f; }

    // SIGMAS / 10; denom = 2*area*(2*sigma)^2 = 8*sigma^2*area
    const float SIG[NUM_KP] = {0.026f, 0.025f, 0.025f, 0.035f, 0.035f, 0.079f,
                               0.079f, 0.072f, 0.072f, 0.062f, 0.062f, 0.107f,
                               0.107f, 0.087f, 0.087f, 0.089f, 0.089f};

    #pragma unroll
    for (int j = 0; j < NUM_KP; ++j) {
        // A fragment (pred row q), lanes l and l+16 read the same LDS word.
        const float px = s_pred[q * 34 + 2 * j];
        const float py = s_pred[q * 34 + 2 * j + 1];
        v2f a, b;
        a.x = half ? py                   : 1.0f;               // K2 | K0
        a.y = half ? (px * px + py * py)  : px;                 // K3 | K1
        const float txj = tx[j], tyj = ty[j];
        b.x = half ? (-2.0f * tyj)        : (txj * txj + tyj * tyj);
        b.y = half ? 1.0f                 : (-2.0f * txj);

        v8f c = {};
        v8f sq = __builtin_amdgcn_wmma_f32_16x16x4_f32(
            /*neg_a=*/false, a, /*neg_b=*/false, b,
            /*c_mod=*/(short)0, c, /*reuse_a=*/false, /*reuse_b=*/false);

        const float cj = 1.0f / (8.0f * SIG[j] * SIG[j]);
        const float sj = cj * inv_area;    // 1 / denom(t, j)
        const float vj = Vv[j];

        #pragma unroll
        for (int i = 0; i < 8; ++i) {
            // OKS accumulation: exp(-sq/denom) * V
            okacc[i] = fmaf(vj, __expf(-sq[i] * sj), okacc[i]);
            // L1 keypoint cost for D row r = i + 8*half (broadcast LDS read).
            const int   r  = i + 8 * half;
            const float gx = s_pred[r * 34 + 2 * j];
            const float gy = s_pred[r * 34 + 2 * j + 1];
            kptacc[i] = fmaf(fabsf(gx - txj) + fabsf(gy - tyj), vj, kptacc[i]);
        }
    }

    #pragma unroll
    for (int i = 0; i < 8; ++i) {
        const int   r   = i + 8 * half;
        const float cc  = s_cc[r * 2 + cls];
        float oks = okacc[i] * inv_vs;
        oks = fmaxf(oks, 1e-6f);
        const float cost = 2.0f * cc + 10.0f * kptacc[i] + 4.0f * (1.0f - oks);
        out[(size_t)(n0 + r) * T + t] = cost;
    }
}

extern "C" void kernel_launch(void* const* d_in, const int* in_sizes, int n_in,
                              void* d_out, int out_size, void* d_ws, size_t ws_size,
                              hipStream_t stream) {
    const float* pred_logits = (const float*)d_in[0];
    // d_in[1] = pred_boxes (unused by reference)
    const float* pred_kp     = (const float*)d_in[2];
    // d_in[3] = tgt_bbox (unused by reference)
    const float* tgt_kp      = (const float*)d_in[4];
    const float* tgt_area    = (const float*)d_in[5];
    const int*   tgt_ids     = (const int*)d_in[6];
    float* out = (float*)d_out;

    const int T = in_sizes[5];          // 320 targets
    const int N = in_sizes[0] / 2;      // nc == 2 -> N = bs*nq = 14400

    dim3 block((T / 16) * 32);          // 20 waves, one 16x16 tile each
    dim3 grid(N / 16);                  // 900 query tiles
    matcher_cost_kernel<<<grid, block, 0, stream>>>(
        pred_logits, pred_kp, tgt_kp, tgt_area, tgt_ids, out, T);
}